// CenterNet_3d_49340584297279
// MI455X (gfx1250) — compile-verified
//
#include <hip/hip_runtime.h>

// Problem constants (from reference): B=32, CAT=1, J=17, H=W=256, K=100
#define B_    32
#define J_    17
#define Hh    256
#define Ww    256
#define HW    65536
#define KK    100
#define KP    112      // K padded to 7*16 for WMMA tiling
#define OUTW  40       // 4 bbox + 1 score + 34 kps + 1 cls
#define THRESH 0.1f

#define ROWS  32                 // NMS stripe height
#define HALO_ROWS (ROWS + 2)     // staged rows incl. halo
#define STAGE_FLOATS (HALO_ROWS * Ww)

typedef __attribute__((ext_vector_type(2))) float v2f;
typedef __attribute__((ext_vector_type(8))) float v8f;
typedef __attribute__((ext_vector_type(4))) unsigned v4u;
typedef __attribute__((ext_vector_type(4))) int v4i;
typedef __attribute__((ext_vector_type(8))) int v8i;

// ---------------------------------------------------------------------------
// TDM helper: issue one Tensor-Data-Mover load of `nrows` x 256 f32 rows
// starting at global row pointer `gsrc` into LDS byte offset `lds_off`.
// D# built per CDNA5 ISA (2D tensor, no padding,
// no gather, no multicast). Caller must s_wait_tensorcnt + barrier.
// ---------------------------------------------------------------------------
#if __has_builtin(__builtin_amdgcn_tensor_load_to_lds)
#define HAVE_TDM 1
__device__ __forceinline__ void tdm_load_rows(const float* gsrc, unsigned lds_off,
                                              int nrows, int rows_remaining) {
  const unsigned long long ga = (unsigned long long)(const void*)gsrc;
  // group0: [1:0]=count=1 | lds_addr | global_addr[56:0] | type=2 ("image")
  v4u g0 = { 1u,
             lds_off,
             (unsigned)(ga & 0xFFFFFFFFu),
             (unsigned)((ga >> 32) & 0x1FFFFFFu) | (2u << 30) };
  // group1:
  //  s0: workgroup_mask=0 | data_size=2 (4B) at [17:16]
  //  s1: atomic_barrier_addr=0 | tensor_dim0[15:0] at [63:48]
  //  s2: tensor_dim0[31:16]=0 | tensor_dim1[15:0] at [111:96->31:16]
  //  s3: tensor_dim1[31:16]=0 | tile_dim0 at [127:112]
  //  s4: tile_dim1 = nrows | tile_dim2 = 0
  //  s5: tensor_dim0_stride low32 = 256
  //  s6: tensor_dim0_stride hi16 = 0 | tensor_dim1_stride low16 = 0
  //  s7: tensor_dim1_stride hi32 = 0 (2D: unused)
  v8i g1 = { (int)(2u << 16),
             (int)((unsigned)Ww << 16),
             (int)(((unsigned)rows_remaining & 0xFFFFu) << 16),
             (int)((unsigned)Ww << 16),
             (int)((unsigned)nrows & 0xFFFFu),
             (int)Ww,
             0,
             0 };
  v4i gz = { 0, 0, 0, 0 };
#if defined(__clang_major__) && __clang_major__ >= 23
  v8i gz8 = { 0, 0, 0, 0, 0, 0, 0, 0 };
  __builtin_amdgcn_tensor_load_to_lds(g0, g1, gz, gz, gz8, 0);
#else
  __builtin_amdgcn_tensor_load_to_lds(g0, g1, gz, gz, 0);
#endif
}
#else
#define HAVE_TDM 0
#endif

// ---------------------------------------------------------------------------
// Kernel 1: per-channel sigmoid + 3x3 NMS + exact top-100.
// One workgroup (256 threads) per channel. Pipeline:
//  (a) stripe loop: TDM async-loads 34 rows (stripe + halo) into an LDS
//      staging tile (s_wait_tensorcnt 0 before use); threads compute the
//      9-point NMS stencil from LDS and write NMS'd sigmoid scores into a
//      channel-sized LDS array (256KB -- needs CDNA5's 320KB/WGP LDS).
//  (b) exact iterative top-100: per-thread (per-column) local max, 100
//      rounds of tree-argmax with tie -> lowest index (jax.lax.top_k order),
//      winner invalidates its element in LDS and rescans its 256 entries.
// NMS keep is decided on raw logits (sigmoid strictly monotonic; keep is
// non-strict: center >= all 8 neighbors, matching hmax == heat).
// ---------------------------------------------------------------------------
__global__ __launch_bounds__(256) void sig_nms_topk(
    const float* __restrict__ heat,
    float* __restrict__ out_scores,
    int*   __restrict__ out_inds) {
  const int ch = blockIdx.x;
  const float* __restrict__ h = heat + (size_t)ch * HW;

  extern __shared__ float sm[];
  float* sc    = sm;                       // HW      : NMS'd scores
  float* lv    = sm + HW;                  // 256     : per-thread local max
  int*   li    = (int*)(sm + HW + 256);    // 256     : per-thread local argmax
  float* rv    = sm + HW + 512;            // 256     : reduce buffer (val)
  int*   ri    = (int*)(sm + HW + 768);    // 256     : reduce buffer (idx)
  float* bc    = sm + HW + 1024;           // 16      : broadcast best val/idx
  float* stage = sm + HW + 1040;           // 34*256  : TDM staging tile

  const int t = threadIdx.x;

  float bestv = -1.0f; int besti = 0;

  // ---- (a) stripe pipeline: TDM -> LDS -> stencil -> score array ----------
  for (int s = 0; s < Hh / ROWS; ++s) {
    const int r0 = s * ROWS;
    const int lo = (r0 == 0) ? 0 : (r0 - 1);                 // first staged row
    const int hi = (r0 + ROWS < Hh) ? (r0 + ROWS) : (Hh - 1); // last staged row
    const int nrows = hi - lo + 1;                            // 33 or 34

#if HAVE_TDM
    if (t < 32) {   // wave 0 issues the DMA; its TENSORcnt gates the barrier
      tdm_load_rows(h + (size_t)lo * Ww,
                    (unsigned)(size_t)(void*)stage,
                    nrows, Hh - lo);
      __builtin_amdgcn_s_wait_tensorcnt(0);
    }
#else
    for (int idx = t; idx < nrows * Ww; idx += 256) {
      stage[idx] = h[(size_t)lo * Ww + idx];
    }
#endif
    __syncthreads();   // stage ready for all waves

    const int c = t;   // this thread owns column c for the whole stripe
    for (int rr = 0; rr < ROWS; ++rr) {
      const int r = r0 + rr;
      const float center = stage[(r - lo) * Ww + c];
      float m = center;
      #pragma unroll
      for (int dr = -1; dr <= 1; ++dr) {
        const int grow = r + dr;
        if (grow < 0 || grow >= Hh) continue;
        const float* srow = stage + (grow - lo) * Ww;
        #pragma unroll
        for (int dc = -1; dc <= 1; ++dc) {
          const int gcol = c + dc;
          if (gcol < 0 || gcol >= Ww) continue;
          m = fmaxf(m, srow[gcol]);
        }
      }
      const float sval = (m == center) ? (1.0f / (1.0f + __expf(-center))) : 0.0f;
      const int e = r * Ww + c;
      sc[e] = sval;
      if (sval > bestv) { bestv = sval; besti = e; }  // scan order => low idx on tie
    }
    __syncthreads();   // stripe consumed before next TDM overwrites stage
  }

  lv[t] = bestv; li[t] = besti;
  __syncthreads();

  // ---- (b) exact iterative top-100 ---------------------------------------
  for (int round = 0; round < KK; ++round) {
    rv[t] = lv[t]; ri[t] = li[t];
    __syncthreads();
    #pragma unroll
    for (int s2 = 128; s2 > 0; s2 >>= 1) {
      if (t < s2) {
        const float a = rv[t], b = rv[t + s2];
        const int ia = ri[t], ib = ri[t + s2];
        if (b > a || (b == a && ib < ia)) { rv[t] = b; ri[t] = ib; }
      }
      __syncthreads();
    }
    if (t == 0) {
      out_scores[ch * KK + round] = rv[0];
      out_inds[ch * KK + round]   = ri[0];
      bc[0] = rv[0];
      ((int*)bc)[1] = ri[0];
    }
    __syncthreads();
    const int bi = ((int*)bc)[1];
    if (t == (bi & 255)) {               // owner (column) of removed element
      sc[bi] = -1.0f;
      float nb = -1.0f; int ni = 0;
      for (int i = 0; i < HW / 256; ++i) {
        const int e = t + 256 * i;
        const float s = sc[e];
        if (s > nb) { nb = s; ni = e; }
      }
      lv[t] = nb; li[t] = ni;
    }
    __syncthreads();
  }
}

// ---------------------------------------------------------------------------
// Kernel 2: per-(b,k) gather reg/wh/hps at top indices; write bbox, score,
// base keypoints (kps_t) and class into the output rows.
// ---------------------------------------------------------------------------
__global__ __launch_bounds__(128) void decode_det(
    const float* __restrict__ wh,  const float* __restrict__ hps,
    const float* __restrict__ reg, const float* __restrict__ tsc,
    const int* __restrict__ tin,   float* __restrict__ out) {
  const int b = blockIdx.x;
  const int k = threadIdx.x;
  if (k >= KK) return;
  const int   ind   = tin[b * KK + k];
  const float score = tsc[b * KK + k];
  const float xsi = (float)(ind & 255);
  const float ysi = (float)(ind >> 8);
  const float* regb = reg + (size_t)b * 2 * HW;
  const float* whb  = wh  + (size_t)b * 2 * HW;
  const float* hpsb = hps + (size_t)b * 2 * J_ * HW;
  const float x = xsi + regb[ind];
  const float y = ysi + regb[HW + ind];
  const float w0 = whb[ind], w1 = whb[HW + ind];
  float* o = out + ((size_t)b * KK + k) * OUTW;
  o[0] = x - w0 * 0.5f;
  o[1] = y - w1 * 0.5f;
  o[2] = x + w0 * 0.5f;
  o[3] = y + w1 * 0.5f;
  o[4] = score;
  #pragma unroll
  for (int j = 0; j < J_; ++j) {
    o[5 + 2 * j]     = hpsb[(size_t)(2 * j) * HW + ind]     + xsi;
    o[5 + 2 * j + 1] = hpsb[(size_t)(2 * j + 1) * HW + ind] + ysi;
  }
  o[39] = 0.0f;   // CAT == 1 -> clses == 0
}

// ---------------------------------------------------------------------------
// Kernel 3: per-(b,j) association. dist^2 = |a|^2 + |b|^2 - 2 a.b; the cross
// term is a 112x2 x 2x112 GEMM done as 7x7 WMMA f32 16x16x4 tiles (K-dim
// lanes 2,3 zero-padded). Then argmin per person-keypoint, exact distance
// recompute for the winner, validity test, conditional overwrite of output.
// ---------------------------------------------------------------------------
__global__ __launch_bounds__(256) void assoc(
    const float* __restrict__ hp_offset,
    const float* __restrict__ hsc_in, const int* __restrict__ hin_in,
    float* __restrict__ out) {
  const int bj = blockIdx.x;
  const int b = bj / J_, j = bj % J_;

  extern __shared__ float sm[];
  float* kx = sm;            // KP  person joint x
  float* ky = kx + KP;       // KP  person joint y
  float* na = ky + KP;       // KP  |a|^2
  float* hx = na + KP;       // KP  heatmap joint x (masked)
  float* hy = hx + KP;       // KP  heatmap joint y (masked)
  float* nb = hy + KP;       // KP  |b|^2 (inf for pad)
  float* hs = nb + KP;       // KP  heatmap joint score (masked)
  float* d2 = hs + KP;       // KP*KP dist^2

  const int t = threadIdx.x;

  if (t < KP) {
    if (t < KK) {
      const float* o = out + ((size_t)b * KK + t) * OUTW;
      const float x = o[5 + 2 * j], y = o[6 + 2 * j];
      kx[t] = x; ky[t] = y; na[t] = x * x + y * y;

      float s = hsc_in[(b * J_ + j) * KK + t];
      const int ind = hin_in[(b * J_ + j) * KK + t];
      const float* offb = hp_offset + (size_t)b * 2 * HW;
      float hxx = (float)(ind & 255) + offb[ind];
      float hyy = (float)(ind >> 8) + offb[HW + ind];
      const bool valid = s > THRESH;
      s   = valid ? s   : -1.0f;
      hxx = valid ? hxx : -10000.0f;
      hyy = valid ? hyy : -10000.0f;
      hx[t] = hxx; hy[t] = hyy; nb[t] = hxx * hxx + hyy * hyy; hs[t] = s;
    } else {
      kx[t] = 0.0f; ky[t] = 0.0f; na[t] = 0.0f;
      hx[t] = 0.0f; hy[t] = 0.0f; nb[t] = 3.0e38f; hs[t] = -1.0f;  // pad never wins
    }
  }
  __syncthreads();

  // WMMA cross terms. Loop bounds are wave-uniform => EXEC all ones at wmma.
  const int lane = t & 31;
  const int wave = t >> 5;          // 8 waves per block
  const int laneHalf = lane >> 4;   // 0: carries K=0,1 ; 1: carries K=2,3 (zeros)
  const int ln = lane & 15;
  for (int tile = wave; tile < 49; tile += 8) {
    const int ti = tile / 7, tj = tile % 7;
    v2f a, bv;
    if (laneHalf == 0) {
      a.x  = kx[ti * 16 + ln]; a.y  = ky[ti * 16 + ln];   // A[m][K=0..1]
      bv.x = hx[tj * 16 + ln]; bv.y = hy[tj * 16 + ln];   // B[K=0..1][n]
    } else {
      a.x = 0.0f; a.y = 0.0f; bv.x = 0.0f; bv.y = 0.0f;   // K=2,3 zero pad
    }
    v8f c = {};
    c = __builtin_amdgcn_wmma_f32_16x16x4_f32(
        /*neg_a=*/false, a, /*neg_b=*/false, bv,
        /*c_mod=*/(short)0, c, /*reuse_a=*/false, /*reuse_b=*/false);
    #pragma unroll
    for (int r = 0; r < 8; ++r) {
      const int m = r + 8 * laneHalf;      // D layout: VGPR r -> M=r / r+8
      const int i = ti * 16 + m;
      const int l = tj * 16 + ln;
      d2[i * KP + l] = na[i] + nb[l] - 2.0f * c[r];
    }
  }
  __syncthreads();

  if (t < KK) {
    const int i = t;
    float mind = 3.4e38f; int mi = 0;
    for (int l = 0; l < KK; ++l) {
      const float v = d2[i * KP + l];
      if (v < mind) { mind = v; mi = l; }   // strict < => first index on tie
    }
    // exact distance for the winner (avoids cancellation at the threshold)
    const float dx = kx[i] - hx[mi], dy = ky[i] - hy[mi];
    const float md = sqrtf(dx * dx + dy * dy);
    const float sx = hx[mi], sy = hy[mi], ss = hs[mi];
    float* o = out + ((size_t)b * KK + i) * OUTW;
    const float l_ = o[0], t_ = o[1], r_ = o[2], b_ = o[3];
    const bool invalid = (sx < l_) || (sx > r_) || (sy < t_) || (sy > b_) ||
                         (ss < THRESH) ||
                         (md > fmaxf(b_ - t_, r_ - l_) * 0.3f);
    if (!invalid) { o[5 + 2 * j] = sx; o[6 + 2 * j] = sy; }
  }
}

// ---------------------------------------------------------------------------
extern "C" void kernel_launch(void* const* d_in, const int* in_sizes, int n_in,
                              void* d_out, int out_size, void* d_ws, size_t ws_size,
                              hipStream_t stream) {
  const float* hm        = (const float*)d_in[0];
  const float* wh        = (const float*)d_in[1];
  const float* hps       = (const float*)d_in[2];
  const float* reg       = (const float*)d_in[3];
  const float* hm_hp     = (const float*)d_in[4];
  const float* hp_offset = (const float*)d_in[5];
  // d_in[6] is K == 100 (compile-time constant here)

  float* out = (float*)d_out;

  char* ws = (char*)d_ws;
  float* tsc = (float*)ws; ws += (size_t)B_ * KK * sizeof(float);
  int*   tin = (int*)ws;   ws += (size_t)B_ * KK * sizeof(int);
  float* hsc = (float*)ws; ws += (size_t)B_ * J_ * KK * sizeof(float);
  int*   hin = (int*)ws;   ws += (size_t)B_ * J_ * KK * sizeof(int);

  const size_t smem1 = (size_t)(HW + 1040 + STAGE_FLOATS) * sizeof(float); // ~294 KB
  sig_nms_topk<<<dim3(B_),       dim3(256), smem1, stream>>>(hm,    tsc, tin);
  sig_nms_topk<<<dim3(B_ * J_),  dim3(256), smem1, stream>>>(hm_hp, hsc, hin);

  decode_det<<<dim3(B_), dim3(128), 0, stream>>>(wh, hps, reg, tsc, tin, out);

  const size_t smem3 = (size_t)(7 * KP + KP * KP) * sizeof(float); // ~52 KB
  assoc<<<dim3(B_ * J_), dim3(256), smem3, stream>>>(hp_offset, hsc, hin, out);
}